// Layered_MS_Decoder_42606075576367
// MI455X (gfx1250) — compile-verified
//
#include <hip/hip_runtime.h>
#include <stdint.h>

// Layered min-sum LDPC decoder for MI455X (gfx1250, wave32).
// All decoder state (vn_llr + c2v) LDS-resident; 16 lanes = 16 edges of one
// check row, 2 codewords per wave; zero barriers (wave-private state);
// async global->LDS staging; leave-one-out min via 4 ds_swizzle SWAPX steps
// on (min1, min2) only; sign parity via one ballot.

#define M_CN   512
#define N_VN   1024
#define W_R    16
#define BATCH  4096

#define EPW     4               // codewords per workgroup
#define THREADS (EPW * W_R)     // 64 threads = 2 waves

template <int PAT>
__device__ __forceinline__ float swzf(float x) {
    return __int_as_float(__builtin_amdgcn_ds_swizzle(__float_as_int(x), PAT));
}

__global__ __launch_bounds__(THREADS)
void ldpc_layered_ms_kernel(const float* __restrict__ chan,     // [B][N]
                            const int*   __restrict__ Hc,       // [M][WR]
                            const int*   __restrict__ iters_p,  // scalar
                            const int*   __restrict__ cn_order, // [M]
                            float*       __restrict__ out)      // [B][N]
{
    // 4*1024 + 4*512*16 floats = 147456 B = 144 KB static LDS
    __shared__ __align__(16) float smem[EPW * N_VN + EPW * M_CN * W_R];
    float* vn  = smem;                  // [EPW][N_VN]
    float* c2v = smem + EPW * N_VN;     // [EPW][M_CN*W_R]

    const int tid = threadIdx.x;
    const int e   = tid & (W_R - 1);    // edge index within check row
    const int bl  = tid >> 4;           // local codeword
    const int b   = blockIdx.x * EPW + bl;
    const unsigned lane       = (unsigned)(tid & 31);
    const unsigned half_shift = lane & 16u;   // which 16-lane half of the wave

    // ---- stage channel LLRs into LDS via gfx1250 async global->LDS (B128) ----
    // per 16-lane group: 4 KB row in 16 chunks of 256 B (16 B per lane, coalesced)
    {
        const float* grow = chan + (size_t)b * N_VN;
        unsigned lds_base = (unsigned)(uintptr_t)(vn + bl * N_VN);
        #pragma unroll
        for (int c = 0; c < 16; ++c) {
            unsigned lds_addr = lds_base + (unsigned)(c * 256 + e * 16);
            unsigned long long gaddr =
                (unsigned long long)(uintptr_t)(grow + c * 64 + e * 4);
            asm volatile("global_load_async_to_lds_b128 %0, %1, off"
                         :: "v"(lds_addr), "v"(gaddr) : "memory");
        }
    }

    // ---- zero this codeword's c2v block (wave-local => no barrier) ----
    {
        float4 z = make_float4(0.f, 0.f, 0.f, 0.f);
        float4* crow = (float4*)(c2v + bl * (M_CN * W_R));
        #pragma unroll 8
        for (int k = 0; k < (M_CN * W_R) / 4 / W_R; ++k)   // 128 b128 stores/lane
            crow[e + k * W_R] = z;
    }

    // wait for this wave's async LDS fills
    asm volatile("s_wait_asynccnt 0" ::: "memory");

    const int iters = iters_p[0];
    float* vrow  = vn  + bl * N_VN;
    float* cbase = c2v + bl * (M_CN * W_R);

    // software prefetch of (cn, column index) one step ahead
    int cn_next  = cn_order[0];
    int idx_next = Hc[cn_next * W_R + e];

    for (int it = 0; it < iters; ++it) {
        for (int s = 0; s < M_CN; ++s) {
            const int cn  = cn_next;
            const int idx = idx_next;
            const int sn  = (s + 1 < M_CN) ? (s + 1) : 0;
            cn_next  = cn_order[sn];            // uniform (s_load, L2-hot)
            idx_next = Hc[cn_next * W_R + e];   // 64B/row, cache-hot

            float v   = vrow[idx];              // LDS gather
            float c   = cbase[cn * W_R + e];    // LDS load
            float v2c = v - c;
            float a   = fabsf(v2c);

            // sign parity of all 16 edges via one ballot (no shuffles)
            unsigned bal  = (unsigned)__ballot(v2c < 0.f);
            int      neg  = (int)((bal >> lane) & 1u);
            unsigned half = (bal >> half_shift) & 0xFFFFu;
            int      par  = __popc(half) & 1;

            // (min1, min2) reduction over 16 lanes: 4 ds_swizzle SWAPX steps.
            // xor-masks 1,2,4,8 never cross the two 16-lane codeword groups.
            float m1 = a, m2 = 3.0e38f;
            #define MS_STEP(P) {                       \
                float om1 = swzf<P>(m1);               \
                float om2 = swzf<P>(m2);               \
                float hi  = fmaxf(m1, om1);            \
                m1 = fminf(m1, om1);                   \
                m2 = fminf(fminf(m2, om2), hi);        \
            }
            MS_STEP(0x041F)   // SWAPX1
            MS_STEP(0x081F)   // SWAPX2
            MS_STEP(0x101F)   // SWAPX4
            MS_STEP(0x201F)   // SWAPX8
            #undef MS_STEP

            // leave-one-out min without argmin: exact under ties, since a tie
            // at the minimum forces m2 == m1.
            float loo = (a == m1) ? m2 : m1;
            loo = fminf(loo, 20.0f);                    // clip(+/-20) on magnitude
            float nc = ((par ^ neg) & 1) ? -loo : loo;  // product of other 15 signs

            vrow[idx]           = v2c + nc;     // LDS scatter (columns distinct)
            cbase[cn * W_R + e] = nc;           // LDS store
        }
    }

    // ---- hard decisions: out = (vn_llr < 0) ? 1 : 0, coalesced b128 stores ----
    float* orow = out + (size_t)b * N_VN;
    #pragma unroll
    for (int c = 0; c < 16; ++c) {
        int off = c * 64 + e * 4;
        float4 vv = *(const float4*)(vrow + off);
        float4 o;
        o.x = (vv.x < 0.f) ? 1.f : 0.f;
        o.y = (vv.y < 0.f) ? 1.f : 0.f;
        o.z = (vv.z < 0.f) ? 1.f : 0.f;
        o.w = (vv.w < 0.f) ? 1.f : 0.f;
        *(float4*)(orow + off) = o;
    }
}

extern "C" void kernel_launch(void* const* d_in, const int* in_sizes, int n_in,
                              void* d_out, int out_size, void* d_ws, size_t ws_size,
                              hipStream_t stream) {
    const float* chan     = (const float*)d_in[0];  // channel_llr [B*N] f32
    const int*   Hc       = (const int*)  d_in[1];  // H_compact  [M*WR] i32
    const int*   iters    = (const int*)  d_in[2];  // scalar i32
    const int*   cn_order = (const int*)  d_in[3];  // [M] i32
    float*       out      = (float*)d_out;          // [B*N] f32

    (void)in_sizes; (void)n_in; (void)out_size; (void)d_ws; (void)ws_size;

    dim3 grid(BATCH / EPW);   // 1024 workgroups
    dim3 block(THREADS);      // 64 threads (2 waves)
    hipLaunchKernelGGL(ldpc_layered_ms_kernel, grid, block, 0, stream,
                       chan, Hc, iters, cn_order, out);
}